// ClassAwareTripletLoss_56521769615603
// MI455X (gfx1250) — compile-verified
//
#include <hip/hip_runtime.h>

typedef __attribute__((ext_vector_type(2))) float v2f;
typedef __attribute__((ext_vector_type(8))) float v8f;
typedef __attribute__((ext_vector_type(4))) unsigned int u32x4;
typedef __attribute__((ext_vector_type(8))) int i32x8;
typedef __attribute__((ext_vector_type(4))) int i32x4;

#define C_CLS 1024
#define D_DIM 64
#define BS    64
#define NROWS (BS * C_CLS)          // 65536
#define LDS_ROW 68                  // 64 floats + 4-dword TDM pad -> conflict-free banks
#define LDS_P2  (C_CLS * LDS_ROW)   // float offset of the p2 mirror in LDS
#define LDS_BYTES ((C_CLS * LDS_ROW + C_CLS) * 4)   // 282624 B <= 320KB WGP LDS

// workspace layout (float offsets)
#define OFF_INV 0                   // 65536 : 1/max(||row||,1e-12)
#define OFF_X2  65536               // 65536 : ||x_norm||^2
#define OFF_SX  131072              // 65536 : sum(x_norm)
#define OFF_P2  196608              // 1024  : ||p||^2
#define OFF_SP  197632              // 1024  : sum(p)
#define OFF_NUM 198656              // 64    : per-batch numerators
#define OFF_DEN 198720              // 64    : per-batch denominators

// ---------------- Kernel 0: per-row and per-prototype statistics ----------------
__global__ void __launch_bounds__(256) tl_prep_kernel(const float* __restrict__ inputs,
                                                      const float* __restrict__ prot,
                                                      float* __restrict__ ws) {
  const int wave = (blockIdx.x * blockDim.x + threadIdx.x) >> 5;
  const int lane = threadIdx.x & 31;
  if (wave < C_CLS) {
    const float* p = prot + wave * D_DIM + lane * 2;
    float a = p[0], b = p[1];
    float p2 = a * a + b * b;
    float sp = a + b;
    for (int m = 16; m; m >>= 1) { p2 += __shfl_xor(p2, m); sp += __shfl_xor(sp, m); }
    if (lane == 0) { ws[OFF_P2 + wave] = p2; ws[OFF_SP + wave] = sp; }
  } else {
    const int r = wave - C_CLS;
    if (r < NROWS) {
      const float* p = inputs + r * D_DIM + lane * 2;
      float a = p[0], b = p[1];
      float n2 = a * a + b * b;
      float s  = a + b;
      for (int m = 16; m; m >>= 1) { n2 += __shfl_xor(n2, m); s += __shfl_xor(s, m); }
      if (lane == 0) {
        float nrm = sqrtf(n2);
        float inv = 1.0f / fmaxf(nrm, 1e-12f);
        float xs  = nrm * inv;
        ws[OFF_INV + r] = inv;
        ws[OFF_X2 + r]  = xs * xs;
        ws[OFF_SX + r]  = s * inv;
      }
    }
  }
}

// ---- one 16x16 output tile: 16 chained f32 WMMAs + branchless argmin scoring ----
template <bool MASKED>
__device__ __forceinline__ void do_tile(int ct, const float* __restrict__ plds,
                                        const v2f* A, int n16, int kOff, int diagJ,
                                        float minS[8], int minIdx[8], float& diagVal) {
  const float* brow = plds + (ct * 16 + n16) * LDS_ROW + kOff;
  v2f B[16];
#pragma unroll
  for (int s = 0; s < 16; ++s) B[s] = *(const v2f*)(brow + 4 * s);

  v8f acc = {0.f, 0.f, 0.f, 0.f, 0.f, 0.f, 0.f, 0.f};
#pragma unroll
  for (int s = 0; s < 16; ++s)
    acc = __builtin_amdgcn_wmma_f32_16x16x4_f32(false, A[s], false, B[s],
                                                (short)0, acc, false, false);

  const int   col = ct * 16 + n16;
  const float p2c = plds[LDS_P2 + col];          // ds_load, no vmem in the hot loop
#pragma unroll
  for (int j = 0; j < 8; ++j) {
    const float sc = fmaf(-2.0f, acc[j], p2c);   // score = p2[k] - 2*xp
    if (MASKED) {
      const bool  isdiag = (j == diagJ);         // k == c -> capture & exclude
      diagVal = isdiag ? acc[j] : diagVal;
      const float scEff = isdiag ? INFINITY : sc;
      const bool  take  = scEff < minS[j];
      minS[j]   = take ? scEff : minS[j];
      minIdx[j] = take ? col   : minIdx[j];
    } else {
      const bool take = sc < minS[j];
      minS[j]   = take ? sc  : minS[j];
      minIdx[j] = take ? col : minIdx[j];
    }
  }
}

// ---------------- Kernel 1: TDM-staged prototypes + fused WMMA GEMM/argmin ----------------
__global__ void __launch_bounds__(256) tl_main_kernel(const float* __restrict__ inputs,
                                                      const float* __restrict__ prot,
                                                      const float* __restrict__ label,
                                                      float* __restrict__ ws) {
  extern __shared__ float plds[];                // [1024 x 68] prot image + [1024] p2 mirror

  const int lane  = threadIdx.x & 31;
  const int wave  = threadIdx.x >> 5;
  // wave-uniform by construction; readfirstlane lets the compiler keep loop
  // bounds / base addresses in SGPRs (scalar loop control instead of exec-mask loops)
  const int strip = __builtin_amdgcn_readfirstlane(blockIdx.x * 8 + wave);
  const int rBase = strip * 16;

  // ---- one TDM descriptor per workgroup: DMA prot (1024x64 f32) -> LDS with +4dw/64dw pad
  if (wave == 0) {
    const unsigned long long pa = (unsigned long long)(uintptr_t)prot;
    const unsigned lbase = (unsigned)(uintptr_t)plds;   // low 32 bits = LDS byte offset
    u32x4 g0;
    g0.x = 1u;                                          // count=1, user descriptor
    g0.y = lbase;                                       // lds_addr
    g0.z = (unsigned)(pa & 0xFFFFFFFFu);                // global_addr[31:0]
    g0.w = (unsigned)((pa >> 32) & 0x01FFFFFFu) | (2u << 30);  // global_addr[56:32] | type=2
    i32x8 g1;
    g1[0] = (int)((2u << 16)        // data_size = 2 -> 4 bytes
                | (1u << 20)        // pad_enable
                | (5u << 22)        // pad_interval = 64 DWORDs
                | (3u << 25));      // pad_amount   = 4 DWORDs
    g1[1] = (int)(64u << 16);       // tensor_dim0 = 64  (bits 63:48)
    g1[2] = (int)(1024u << 16);     // tensor_dim1 = 1024 (bits 95:80)
    g1[3] = (int)(64u << 16);       // tile_dim0 = 64 (bits 127:112)
    g1[4] = (int)(1024u);           // tile_dim1 = 1024, tile_dim2 = 0
    g1[5] = 64;                     // tensor_dim0_stride = 64
    g1[6] = 0;
    g1[7] = 0;
    const i32x4 z4 = {0, 0, 0, 0};              // groups 2/3 unused (2-D tensor)
    const i32x8 z8 = {0, 0, 0, 0, 0, 0, 0, 0};  // extra operand (unused), zeroed
    __builtin_amdgcn_tensor_load_to_lds(g0, g1, z4, z4, z8, 0);
  }

  // ---- overlap with the DMA: mirror p2 into the LDS tail (disjoint region)
  for (int i = threadIdx.x; i < C_CLS; i += 256) plds[LDS_P2 + i] = ws[OFF_P2 + i];

  const int hi   = lane >> 4;                    // ISA: lanes 16-31 hold K offset +2
  const int n16  = lane & 15;
  const int kOff = hi * 2;
  const int diagJ = (lane < 8) ? lane : ((lane >= 24) ? (lane - 24) : -1);

  // ---- A matrix (global, independent of LDS): 16 rows x K=64 f32, prescaled by 1/||x||
  const float* arow = inputs + (size_t)(rBase + n16) * D_DIM + kOff;
  const float  invn = ws[OFF_INV + rBase + n16];
  v2f A[16];
#pragma unroll
  for (int s = 0; s < 16; ++s) {
    v2f t = *(const v2f*)(arow + 4 * s);         // K = 4s + kOff + {0,1}
    A[s] = t * invn;
  }

  if (wave == 0) __builtin_amdgcn_s_wait_tensorcnt(0);
  __syncthreads();

  float minS[8];
  int   minIdx[8];
#pragma unroll
  for (int j = 0; j < 8; ++j) { minS[j] = INFINITY; minIdx[j] = 0; }
  float diagVal = 0.0f;
  const int diagTile = __builtin_amdgcn_readfirstlane((rBase & (C_CLS - 1)) >> 4);

  // uniform split: fast tiles / one masked tile / fast tiles (ascending ct keeps tie order)
  for (int ct = 0; ct < diagTile; ++ct)
    do_tile<false>(ct, plds, A, n16, kOff, diagJ, minS, minIdx, diagVal);
  do_tile<true>(diagTile, plds, A, n16, kOff, diagJ, minS, minIdx, diagVal);
  for (int ct = diagTile + 1; ct < 64; ++ct)
    do_tile<false>(ct, plds, A, n16, kOff, diagJ, minS, minIdx, diagVal);

  // ---- cross-lane argmin within each 16-lane half (rows 0-7 | rows 8-15), branchless
#pragma unroll
  for (int j = 0; j < 8; ++j) {
    float s   = minS[j];
    int   idx = minIdx[j];
    for (int m = 1; m <= 8; m <<= 1) {
      const float os   = __shfl_xor(s, m);
      const int   oi   = __shfl_xor(idx, m);
      const bool  take = (os < s) || (os == s && oi < idx);
      s   = take ? os : s;
      idx = take ? oi : idx;
    }
    minS[j] = s; minIdx[j] = idx;
  }

  // ---- epilogue on the diagonal-owning lanes (one per row)
  float triw = 0.0f, wv = 0.0f;
  const bool writer = (lane < 8) || (lane >= 24);
  if (writer) {
    const int j   = lane & 7;
    const int row = (lane < 8) ? lane : (lane - 16);
    const int r   = rBase + row;
    const int c   = r & (C_CLS - 1);
    const int k   = minIdx[j];
    const float x2  = ws[OFF_X2 + r];
    const float sx  = ws[OFF_SX + r];
    const float p2c = ws[OFF_P2 + c];
    const float spc = ws[OFF_SP + c];
    const float spk = ws[OFF_SP + k];
    const float e   = 1e-6f;
    const float De2 = (float)D_DIM * 1e-12f;
    // ||x - p + e||^2 = x2 + p2 + D*e^2 - 2*x.p + 2e*(sum(x) - sum(p))
    const float ap2 = x2 + p2c - 2.0f * diagVal + De2 + 2.0f * e * (sx - spc);
    const float an2 = x2 + minS[j]              + De2 + 2.0f * e * (sx - spk);
    const float dap = sqrtf(fmaxf(ap2, 0.0f));
    const float dan = sqrtf(fmaxf(an2, 0.0f));
    const float tri = fmaxf(dap - dan + 0.2f, 0.0f);
    const float w   = label[r];
    triw = tri * w;
    wv   = w;
  }
  for (int m = 16; m; m >>= 1) { triw += __shfl_xor(triw, m); wv += __shfl_xor(wv, m); }
  if (lane == 0) {
    const int b = rBase >> 10;
    atomicAdd(&ws[OFF_NUM + b], triw);
    atomicAdd(&ws[OFF_DEN + b], wv);
  }
}

// ---------------- Kernel 2: loss = mean_b( numer[b] / denom[b] ) ----------------
__global__ void tl_finish_kernel(const float* __restrict__ ws, float* __restrict__ out) {
  __shared__ float sh[BS];
  const int t = threadIdx.x;
  sh[t] = ws[OFF_NUM + t] / ws[OFF_DEN + t];
  __syncthreads();
  if (t == 0) {
    float s = 0.0f;
    for (int i = 0; i < BS; ++i) s += sh[i];
    out[0] = s * (1.0f / (float)BS);
  }
}

extern "C" void kernel_launch(void* const* d_in, const int* in_sizes, int n_in,
                              void* d_out, int out_size, void* d_ws, size_t ws_size,
                              hipStream_t stream) {
  const float* inputs = (const float*)d_in[0];   // (64,1024,64) fp32
  const float* label  = (const float*)d_in[1];   // (64,1024,1)  fp32
  const float* prot   = (const float*)d_in[2];   // (1024,64)    fp32
  float* ws = (float*)d_ws;

  // allow >64KB dynamic LDS for the prototype image (deterministic, capture-safe host call)
  (void)hipFuncSetAttribute((const void*)tl_main_kernel,
                            hipFuncAttributeMaxDynamicSharedMemorySize, LDS_BYTES);

  // zero the per-batch accumulators (graph-capture safe)
  (void)hipMemsetAsync(ws + OFF_NUM, 0, 128 * sizeof(float), stream);

  tl_prep_kernel<<<8320, 256, 0, stream>>>(inputs, prot, ws);
  tl_main_kernel<<<512, 256, LDS_BYTES, stream>>>(inputs, prot, label, ws);
  tl_finish_kernel<<<1, BS, 0, stream>>>(ws, (float*)d_out);
}